// RefMLPEncoder_12275016532223
// MI455X (gfx1250) — compile-verified
//
#include <hip/hip_runtime.h>
#include <math.h>

typedef __attribute__((ext_vector_type(2))) float v2f;
typedef __attribute__((ext_vector_type(8))) float v8f;

#define B_   64
#define T_   10
#define N_   64
#define D_   4
#define H_   128
#define E_   (N_ * (N_ - 1))      // 4032
#define ME_  (B_ * E_)            // 258048 edge rows
#define MN_  (B_ * N_)            // 4096 node rows
#define LDSS 132                  // padded LDS row stride (bank-conflict free)

// ---------------------------------------------------------------------------
// [B,T,N,D] -> [B*N, T*D]
// ---------------------------------------------------------------------------
__global__ void transpose_kernel(const float* __restrict__ in, float* __restrict__ out) {
    int idx = blockIdx.x * blockDim.x + threadIdx.x;      // over B*N*T*D = 163840
    if (idx >= B_ * N_ * T_ * D_) return;
    int d = idx & 3;
    int t = (idx / 4) % T_;
    int n = (idx / (T_ * D_)) % N_;
    int b = idx / (T_ * D_ * N_);
    out[idx] = in[(((size_t)b * T_ + t) * N_ + n) * D_ + d];
}

// ---------------------------------------------------------------------------
// Repack weight matrix W[K,128] into B-fragment pair order:
//   Wp[(k/2)*128 + n] = { W[k,n], W[k+1,n] }  (stored as float2)
// so each lane's B-fragment is a single 8-byte load in the GEMM hot loop.
// ---------------------------------------------------------------------------
__global__ void repack_kernel(const float* __restrict__ w, float* __restrict__ wp, int K) {
    int idx = blockIdx.x * blockDim.x + threadIdx.x;      // over (K/2)*128
    if (idx >= (K / 2) * H_) return;
    int n = idx & (H_ - 1);
    int p = idx >> 7;
    wp[idx * 2]     = w[(size_t)(2 * p) * H_ + n];
    wp[idx * 2 + 1] = w[(size_t)(2 * p + 1) * H_ + n];
}

// ---------------------------------------------------------------------------
// Fused 2-layer MLP: out = ELU( ELU(A @ w1 + b1) @ w2 + b2 )
// One wave (32 threads) per 16-row M-tile; hidden tile staged through LDS.
// MODE 0: A = src0[row, 0..K)                       (plain, K = 40 or 128)
// MODE 1: A = concat(x[b,SEND[e]], x[b,RECV[e]])    (K = 256, src0 = node emb)
// MODE 2: A = concat(x3[b,SEND], x3[b,RECV], skip)  (K = 384, src1 = x_skip)
// w1p/w2p are pair-repacked weights (see repack_kernel).
// WMMA: V_WMMA_F32_16X16X4_F32, fp32 end to end.
// ---------------------------------------------------------------------------
template <int MODE>
__global__ __launch_bounds__(32) void mlp_fused_kernel(
    const float* __restrict__ src0, const float* __restrict__ src1,
    const v2f* __restrict__ w1p, const float* __restrict__ b1,
    const v2f* __restrict__ w2p, const float* __restrict__ b2,
    float* __restrict__ out, int K)
{
    __shared__ float lds[16 * LDSS];

    const int lane  = threadIdx.x;          // 0..31
    const int mloc  = lane & 15;            // M (A) or N (B/C) within tile
    const int khalf = (lane >> 4) * 2;      // K sub-offset: 0 or 2
    const int hi    = (lane >> 4) * 8;      // C/D row offset for upper half-wave
    const size_t row = (size_t)blockIdx.x * 16 + mloc;

    // Per-lane A-row source pointers (fused node2edge gather).
    const float* pS = nullptr;   // send-node features
    const float* pR = nullptr;   // recv-node features
    const float* pK = nullptr;   // skip features
    if (MODE == 0) {
        pS = src0 + row * (size_t)K;
    } else {
        int b  = (int)(row / E_);
        int e  = (int)(row % E_);
        int i  = e / (N_ - 1);
        int jj = e % (N_ - 1);
        int j  = jj + (jj >= i ? 1 : 0);
        pS = src0 + ((size_t)(b * N_ + i)) * H_;
        pR = src0 + ((size_t)(b * N_ + j)) * H_;
        if (MODE == 2) pK = src1 + row * (size_t)H_;
    }

    v8f acc[8];
#pragma unroll
    for (int t = 0; t < 8; ++t)
#pragma unroll
        for (int r = 0; r < 8; ++r) acc[t][r] = 0.0f;

    // ---- layer 1: acc = A @ w1 ----
    for (int k0 = 0; k0 < K; k0 += 4) {
        const int k = k0 + khalf;           // even; 8-byte aligned accesses below
        v2f a;
        if (MODE == 0) {
            a = *(const v2f*)(pS + k);
        } else if (MODE == 1) {
            const float* p = (k < H_) ? pS : pR;
            const int kk   = (k < H_) ? k : k - H_;
            a = *(const v2f*)(p + kk);
        } else {
            const float* p; int kk;
            if (k < H_)          { p = pS; kk = k; }
            else if (k < 2 * H_) { p = pR; kk = k - H_; }
            else                 { p = pK; kk = k - 2 * H_; }
            a = *(const v2f*)(p + kk);
        }
        const v2f* wr = w1p + (size_t)(k >> 1) * H_;
#pragma unroll
        for (int t = 0; t < 8; ++t) {
            const v2f bf = wr[t * 16 + mloc];           // single b64 load
            acc[t] = __builtin_amdgcn_wmma_f32_16x16x4_f32(
                false, a, false, bf, (short)0, acc[t], false, false);
        }
    }

    // ---- bias + ELU, stage hidden tile in LDS (C-layout -> row-major) ----
#pragma unroll
    for (int t = 0; t < 8; ++t) {
        const int n = t * 16 + mloc;
        const float bb = b1[n];
#pragma unroll
        for (int r = 0; r < 8; ++r) {
            float h = acc[t][r] + bb;
            h = (h > 0.0f) ? h : expm1f(h);
            lds[(r + hi) * LDSS + n] = h;
            acc[t][r] = 0.0f;
        }
    }

    // ---- layer 2: acc = hidden @ w2 (A-frags from LDS, K = 128) ----
    for (int k0 = 0; k0 < H_; k0 += 4) {
        const int k = k0 + khalf;
        const v2f a = *(const v2f*)(&lds[mloc * LDSS + k]);   // ds_load_b64
        const v2f* wr = w2p + (size_t)(k >> 1) * H_;
#pragma unroll
        for (int t = 0; t < 8; ++t) {
            const v2f bf = wr[t * 16 + mloc];
            acc[t] = __builtin_amdgcn_wmma_f32_16x16x4_f32(
                false, a, false, bf, (short)0, acc[t], false, false);
        }
    }

    // ---- bias + ELU, store pre-BN output tile ----
    const size_t rowbase = (size_t)blockIdx.x * 16;
#pragma unroll
    for (int t = 0; t < 8; ++t) {
        const int n = t * 16 + mloc;
        const float bb = b2[n];
#pragma unroll
        for (int r = 0; r < 8; ++r) {
            float v = acc[t][r] + bb;
            v = (v > 0.0f) ? v : expm1f(v);
            out[(rowbase + r + hi) * H_ + n] = v;
        }
    }
}

// ---------------------------------------------------------------------------
// BatchNorm stats: per-channel sum / sumsq with per-block accumulation.
// stats[0..127] = sum, stats[128..255] = sumsq (zeroed before launch).
// ---------------------------------------------------------------------------
__global__ __launch_bounds__(128) void bn_stats_kernel(
    const float* __restrict__ x, int M, float* __restrict__ stats)
{
    const int c = threadIdx.x;                 // channel
    size_t r0 = (size_t)blockIdx.x * 256;
    size_t r1 = r0 + 256; if (r1 > (size_t)M) r1 = (size_t)M;
    float s = 0.0f, s2 = 0.0f;
    for (size_t r = r0; r < r1; ++r) {
        float v = x[r * H_ + c];
        s += v; s2 += v * v;
    }
    atomicAdd(&stats[c], s);
    atomicAdd(&stats[H_ + c], s2);
}

__global__ void bn_apply_kernel(float* __restrict__ x, int M,
                                const float* __restrict__ stats,
                                const float* __restrict__ g,
                                const float* __restrict__ beta, float invM)
{
    const size_t total = (size_t)M * H_;
    for (size_t idx = (size_t)blockIdx.x * blockDim.x + threadIdx.x; idx < total;
         idx += (size_t)gridDim.x * blockDim.x) {
        int c = (int)(idx & (H_ - 1));
        float mu  = stats[c] * invM;
        float var = stats[H_ + c] * invM - mu * mu;
        float inv = rsqrtf(var + 1e-5f);
        x[idx] = (x[idx] - mu) * inv * g[c] + beta[c];
    }
}

// ---------------------------------------------------------------------------
// edge2node: Xagg[b,n,h] = (1/(N-1)) * sum_{i != n} X2[b, e(i,n), h]
// e(i,n) = i*(N-1) + (n - (n > i))
// ---------------------------------------------------------------------------
__global__ __launch_bounds__(128) void edge2node_kernel(
    const float* __restrict__ x2, float* __restrict__ agg)
{
    const int h = threadIdx.x;
    const int b = blockIdx.x / N_;
    const int n = blockIdx.x % N_;
    float s = 0.0f;
    for (int i = 0; i < N_; ++i) {
        if (i == n) continue;
        int e = i * (N_ - 1) + (n - (n > i ? 1 : 0));
        s += x2[((size_t)b * E_ + e) * H_ + h];
    }
    agg[((size_t)b * N_ + n) * H_ + h] = s * (1.0f / (N_ - 1));
}

// ---------------------------------------------------------------------------
// fc: out[m, 0..1] = x[m, :] @ fc_w + fc_b   (fc_w is [128, 2] row-major)
// ---------------------------------------------------------------------------
__global__ void fc_kernel(const float* __restrict__ x, const float* __restrict__ w,
                          const float* __restrict__ b, float* __restrict__ out, int M)
{
    int m = blockIdx.x * blockDim.x + threadIdx.x;
    if (m >= M) return;
    const float* xr = x + (size_t)m * H_;
    float a0 = b[0], a1 = b[1];
#pragma unroll 8
    for (int h = 0; h < H_; ++h) {
        float v = xr[h];
        a0 += v * w[h * 2];
        a1 += v * w[h * 2 + 1];
    }
    out[(size_t)m * 2]     = a0;
    out[(size_t)m * 2 + 1] = a1;
}

// ---------------------------------------------------------------------------
extern "C" void kernel_launch(void* const* d_in, const int* in_sizes, int n_in,
                              void* d_out, int out_size, void* d_ws, size_t ws_size,
                              hipStream_t stream)
{
    const float* inp = (const float*)d_in[0];
    const float* m1[6]; const float* m2[6]; const float* m3[6]; const float* m4[6];
    for (int q = 0; q < 6; ++q) {
        m1[q] = (const float*)d_in[1 + q];
        m2[q] = (const float*)d_in[7 + q];
        m3[q] = (const float*)d_in[13 + q];
        m4[q] = (const float*)d_in[19 + q];
    }
    const float* fcw = (const float*)d_in[25];
    const float* fcb = (const float*)d_in[26];
    float* outp = (float*)d_out;

    // Workspace layout (floats)
    float* ws    = (float*)d_ws;
    float* X0    = ws;                               // 4096*40
    float* X1    = X0 + (size_t)MN_ * (T_ * D_);     // 4096*128
    float* X2    = X1 + (size_t)MN_ * H_;            // 258048*128 (x_skip)
    float* X3    = X2 + (size_t)ME_ * H_;            // 4096*128
    float* Xagg  = X3 + (size_t)MN_ * H_;            // 4096*128
    float* X4    = Xagg + (size_t)MN_ * H_;          // 258048*128
    float* stats = X4 + (size_t)ME_ * H_;            // 256
    // pair-repacked weights (float2 element order, sizes in floats)
    float* w1p1  = stats + 256;                      // 40*128
    float* w2p1  = w1p1 + (size_t)(T_ * D_) * H_;    // 128*128
    float* w1p2  = w2p1 + (size_t)H_ * H_;           // 256*128
    float* w2p2  = w1p2 + (size_t)(2 * H_) * H_;     // 128*128
    float* w1p3  = w2p2 + (size_t)H_ * H_;           // 128*128
    float* w2p3  = w1p3 + (size_t)H_ * H_;           // 128*128
    float* w1p4  = w2p3 + (size_t)H_ * H_;           // 384*128
    float* w2p4  = w1p4 + (size_t)(3 * H_) * H_;     // 128*128

    const int statsBytes = 2 * H_ * (int)sizeof(float);

    // 0) repack all weight matrices into B-fragment pair order
    {
        struct { const float* w; float* wp; int K; } rp[8] = {
            { m1[0], w1p1, T_ * D_ }, { m1[2], w2p1, H_ },
            { m2[0], w1p2, 2 * H_ },  { m2[2], w2p2, H_ },
            { m3[0], w1p3, H_ },      { m3[2], w2p3, H_ },
            { m4[0], w1p4, 3 * H_ },  { m4[2], w2p4, H_ },
        };
        for (int q = 0; q < 8; ++q) {
            int tot = (rp[q].K / 2) * H_;
            repack_kernel<<<(tot + 255) / 256, 256, 0, stream>>>(rp[q].w, rp[q].wp, rp[q].K);
        }
    }

    // 1) transpose inputs -> X0 [4096, 40]
    transpose_kernel<<<(B_ * N_ * T_ * D_ + 255) / 256, 256, 0, stream>>>(inp, X0);

    // 2) mlp1 (K=40) -> X1, then BN
    mlp_fused_kernel<0><<<MN_ / 16, 32, 0, stream>>>(X0, nullptr,
        (const v2f*)w1p1, m1[1], (const v2f*)w2p1, m1[3], X1, T_ * D_);
    hipMemsetAsync(stats, 0, statsBytes, stream);
    bn_stats_kernel<<<MN_ / 256, 128, 0, stream>>>(X1, MN_, stats);
    bn_apply_kernel<<<512, 256, 0, stream>>>(X1, MN_, stats, m1[4], m1[5], 1.0f / MN_);

    // 3) mlp2 (gathered K=256) -> X2, then BN; X2 becomes x_skip
    mlp_fused_kernel<1><<<ME_ / 16, 32, 0, stream>>>(X1, nullptr,
        (const v2f*)w1p2, m2[1], (const v2f*)w2p2, m2[3], X2, 2 * H_);
    hipMemsetAsync(stats, 0, statsBytes, stream);
    bn_stats_kernel<<<ME_ / 256, 128, 0, stream>>>(X2, ME_, stats);
    bn_apply_kernel<<<4096, 256, 0, stream>>>(X2, ME_, stats, m2[4], m2[5], 1.0f / ME_);

    // 4) edge2node aggregation -> Xagg [4096, 128]
    edge2node_kernel<<<B_ * N_, 128, 0, stream>>>(X2, Xagg);

    // 5) mlp3 (K=128) -> X3, then BN
    mlp_fused_kernel<0><<<MN_ / 16, 32, 0, stream>>>(Xagg, nullptr,
        (const v2f*)w1p3, m3[1], (const v2f*)w2p3, m3[3], X3, H_);
    hipMemsetAsync(stats, 0, statsBytes, stream);
    bn_stats_kernel<<<MN_ / 256, 128, 0, stream>>>(X3, MN_, stats);
    bn_apply_kernel<<<512, 256, 0, stream>>>(X3, MN_, stats, m3[4], m3[5], 1.0f / MN_);

    // 6) mlp4 (gathered K=384 with skip) -> X4, then BN
    mlp_fused_kernel<2><<<ME_ / 16, 32, 0, stream>>>(X3, X2,
        (const v2f*)w1p4, m4[1], (const v2f*)w2p4, m4[3], X4, 3 * H_);
    hipMemsetAsync(stats, 0, statsBytes, stream);
    bn_stats_kernel<<<ME_ / 256, 128, 0, stream>>>(X4, ME_, stats);
    bn_apply_kernel<<<4096, 256, 0, stream>>>(X4, ME_, stats, m4[4], m4[5], 1.0f / ME_);

    // 7) final fc -> logits [B, E, 2]
    fc_kernel<<<(ME_ + 255) / 256, 256, 0, stream>>>(X4, fcw, fcb, outp, ME_);
}